// EProPnPeasy_45672682225704
// MI455X (gfx1250) — compile-verified
//
#include <hip/hip_runtime.h>
#include <hip/hip_bf16.h>
#include <math.h>

typedef __attribute__((ext_vector_type(2))) float v2f;
typedef __attribute__((ext_vector_type(8))) float v8f;

#define OBJ 64
#define NPTS 512
#define MC 512
#define Z_MIN 1e-4f
#define EPSV 1e-5f

// d_out layout (floats)
#define POSE_OFF 0                       // 512*64*7 = 229376
#define LW_OFF   229376                  // 512*64   = 32768
#define CI_OFF   262144                  // 64

// ws layout: per object, stride 40 floats:
// [0:3) offset  [3:6) t0  [6:15) L_t row-major  [15:31) L_r row-major  [31] hld_t  [32] hld_r
#define WSTRIDE 40

#define C_LP_T 2.7568155996140186f   // 1.5 * log(2*pi)
#define C_LP_R 2.9826069522913796f   // log(2*pi^2)

__device__ __forceinline__ float block_reduce_sum(float v, float* red) {
  const int tid = threadIdx.x;
  red[tid] = v;
  __syncthreads();
  for (int s = 128; s > 0; s >>= 1) {
    if (tid < s) red[tid] += red[tid + s];
    __syncthreads();
  }
  float r = red[0];
  __syncthreads();
  return r;
}

__device__ __forceinline__ void quat_rot(float qw, float qi, float qj, float qk, float R[3][3]) {
  R[0][0] = 1.f - 2.f * (qj * qj + qk * qk);
  R[0][1] = 2.f * (qi * qj - qk * qw);
  R[0][2] = 2.f * (qi * qk + qj * qw);
  R[1][0] = 2.f * (qi * qj + qk * qw);
  R[1][1] = 1.f - 2.f * (qi * qi + qk * qk);
  R[1][2] = 2.f * (qj * qk - qi * qw);
  R[2][0] = 2.f * (qi * qk - qj * qw);
  R[2][1] = 2.f * (qj * qk + qi * qw);
  R[2][2] = 1.f - 2.f * (qi * qi + qj * qj);
}

// ---------------- Kernel 1: per-object setup + cost_init ----------------
__global__ void epropnp_setup_kernel(const float* __restrict__ x3d,
                                     const float* __restrict__ x2d,
                                     const float* __restrict__ w2d,
                                     const float* __restrict__ pose,
                                     const float* __restrict__ cam,
                                     const float* __restrict__ cov_t,
                                     const float* __restrict__ cov_r,
                                     float* __restrict__ ws,
                                     float* __restrict__ out) {
  const int o = blockIdx.x;
  const int tid = threadIdx.x;
  __shared__ float red[256];
  __shared__ float sh[16];  // [0:3) offset, [3:12) M0, [12:15) b0

  // ---- offset = mean(x3d) over points ----
  float sx = 0.f, sy = 0.f, sz = 0.f;
  for (int p = tid; p < NPTS; p += 256) {
    const float* xp = x3d + ((size_t)o * NPTS + p) * 3;
    sx += xp[0]; sy += xp[1]; sz += xp[2];
  }
  float offx = block_reduce_sum(sx, red) * (1.f / NPTS);
  float offy = block_reduce_sum(sy, red) * (1.f / NPTS);
  float offz = block_reduce_sum(sz, red) * (1.f / NPTS);

  if (tid == 0) {
    float* W = ws + (size_t)o * WSTRIDE;
    const float* P = pose + o * 7;
    float qw = P[3], qi = P[4], qj = P[5], qk = P[6];
    float R0[3][3];
    quat_rot(qw, qi, qj, qk, R0);
    float t0x = P[0] + R0[0][0] * offx + R0[0][1] * offy + R0[0][2] * offz;
    float t0y = P[1] + R0[1][0] * offx + R0[1][1] * offy + R0[1][2] * offz;
    float t0z = P[2] + R0[2][0] * offx + R0[2][1] * offy + R0[2][2] * offz;

    // Cholesky 3x3
    const float* A = cov_t + o * 9;
    float L00 = sqrtf(A[0]);
    float L10 = A[3] / L00;
    float L20 = A[6] / L00;
    float L11 = sqrtf(A[4] - L10 * L10);
    float L21 = (A[7] - L20 * L10) / L11;
    float L22 = sqrtf(A[8] - L20 * L20 - L21 * L21);

    // Cholesky 4x4
    const float* B = cov_r + o * 16;
    float N00 = sqrtf(B[0]);
    float N10 = B[4] / N00, N20 = B[8] / N00, N30 = B[12] / N00;
    float N11 = sqrtf(B[5] - N10 * N10);
    float N21 = (B[9] - N20 * N10) / N11;
    float N31 = (B[13] - N30 * N10) / N11;
    float N22 = sqrtf(B[10] - N20 * N20 - N21 * N21);
    float N32 = (B[14] - N30 * N20 - N31 * N21) / N22;
    float N33 = sqrtf(B[15] - N30 * N30 - N31 * N31 - N32 * N32);

    W[0] = offx; W[1] = offy; W[2] = offz;
    W[3] = t0x;  W[4] = t0y;  W[5] = t0z;
    W[6] = L00; W[7] = 0.f; W[8] = 0.f;
    W[9] = L10; W[10] = L11; W[11] = 0.f;
    W[12] = L20; W[13] = L21; W[14] = L22;
    float* Lr = W + 15;
    Lr[0] = N00; Lr[1] = 0.f; Lr[2] = 0.f; Lr[3] = 0.f;
    Lr[4] = N10; Lr[5] = N11; Lr[6] = 0.f; Lr[7] = 0.f;
    Lr[8] = N20; Lr[9] = N21; Lr[10] = N22; Lr[11] = 0.f;
    Lr[12] = N30; Lr[13] = N31; Lr[14] = N32; Lr[15] = N33;
    W[31] = logf(L00) + logf(L11) + logf(L22);
    W[32] = logf(N00) + logf(N11) + logf(N22) + logf(N33);

    // M0 = K*R0, b0 = K*t0 (for cost_init)
    const float* Kp = cam + o * 9;
    float ts[3] = {t0x, t0y, t0z};
    for (int i = 0; i < 3; i++) {
      for (int j = 0; j < 3; j++)
        sh[3 + i * 3 + j] = Kp[i * 3 + 0] * R0[0][j] + Kp[i * 3 + 1] * R0[1][j] + Kp[i * 3 + 2] * R0[2][j];
      sh[12 + i] = Kp[i * 3 + 0] * ts[0] + Kp[i * 3 + 1] * ts[1] + Kp[i * 3 + 2] * ts[2];
    }
    sh[0] = offx; sh[1] = offy; sh[2] = offz;
  }
  __syncthreads();

  // ---- cost_init ----
  float c = 0.f;
  for (int p = tid; p < NPTS; p += 256) {
    const float* xp = x3d + ((size_t)o * NPTS + p) * 3;
    float px = xp[0] - sh[0], py = xp[1] - sh[1], pz = xp[2] - sh[2];
    float u = sh[3] * px + sh[4] * py + sh[5] * pz + sh[12];
    float v = sh[6] * px + sh[7] * py + sh[8] * pz + sh[13];
    float w = sh[9] * px + sh[10] * py + sh[11] * pz + sh[14];
    float z = fmaxf(w, Z_MIN);
    float iz = __builtin_amdgcn_rcpf(z);   // z >= 1e-4, hardware rcp (1 ulp) is safe
    const float* x2p = x2d + ((size_t)o * NPTS + p) * 2;
    const float* w2p = w2d + ((size_t)o * NPTS + p) * 2;
    float ru = (u * iz - x2p[0]) * w2p[0];
    float rv = (v * iz - x2p[1]) * w2p[1];
    c = fmaf(ru, ru, fmaf(rv, rv, c));
  }
  float total = block_reduce_sum(c, red);
  if (tid == 0) out[CI_OFF + o] = 0.5f * total;
}

// ---------------- Kernel 2: per (mc-tile, obj) sampling + WMMA cost ----------------
__global__ void epropnp_sample_kernel(const float* __restrict__ x3d,
                                      const float* __restrict__ x2d,
                                      const float* __restrict__ w2d,
                                      const float* __restrict__ cam,
                                      const float* __restrict__ noise_t,
                                      const float* __restrict__ noise_r,
                                      const float* __restrict__ ws,
                                      float* __restrict__ out) {
  const int tile = blockIdx.x;  // 0..31 (MC tiles of 16)
  const int o = blockIdx.y;     // 0..63
  const int tid = threadIdx.x;  // 256 threads = 8 wave32

  __shared__ float As[16][16];       // per sample: [Mrow0 b0 | Mrow1 b1 | Mrow2 b2 | lpsum pad3]
  __shared__ float wave_part[8][16]; // per-wave per-sample partial costs

  const float* W = ws + (size_t)o * WSTRIDE;
  const float offx = W[0], offy = W[1], offz = W[2];

  if (tid < 16) {
    const int s = tid;
    const int mc = tile * 16 + s;
    const float* nt = noise_t + ((size_t)mc * OBJ + o) * 3;
    const float* nr = noise_r + ((size_t)mc * OBJ + o) * 4;
    const float n0 = nt[0], n1 = nt[1], n2 = nt[2];
    const float r0 = nr[0], r1 = nr[1], r2 = nr[2], r3 = nr[3];
    const float* Lt = W + 6;
    const float* Lr = W + 15;

    float tsx = W[3] + Lt[0] * n0;
    float tsy = W[4] + Lt[3] * n0 + Lt[4] * n1;
    float tsz = W[5] + Lt[6] * n0 + Lt[7] * n1 + Lt[8] * n2;

    float g0 = Lr[0] * r0;
    float g1 = Lr[4] * r0 + Lr[5] * r1;
    float g2 = Lr[8] * r0 + Lr[9] * r1 + Lr[10] * r2;
    float g3 = Lr[12] * r0 + Lr[13] * r1 + Lr[14] * r2 + Lr[15] * r3;
    float gn = sqrtf(g0 * g0 + g1 * g1 + g2 * g2 + g3 * g3);

    float qw, qi, qj, qk, Mr;
    if (gn < EPSV) {
      qw = 1.f; qi = 0.f; qj = 0.f; qk = 0.f;
      // forward-solve L_r y = unit
      float y0 = 1.f / Lr[0];
      float y1 = -(Lr[4] * y0) / Lr[5];
      float y2 = -(Lr[8] * y0 + Lr[9] * y1) / Lr[10];
      float y3 = -(Lr[12] * y0 + Lr[13] * y1 + Lr[14] * y2) / Lr[15];
      Mr = y0 * y0 + y1 * y1 + y2 * y2 + y3 * y3;
    } else {
      float inv = 1.f / gn;
      qw = g0 * inv; qi = g1 * inv; qj = g2 * inv; qk = g3 * inv;
      Mr = (r0 * r0 + r1 * r1 + r2 * r2 + r3 * r3) * inv * inv;
    }

    float R[3][3];
    quat_rot(qw, qi, qj, qk, R);

    const float* Kp = cam + o * 9;
    float ts[3] = {tsx, tsy, tsz};
    #pragma unroll
    for (int i = 0; i < 3; i++) {
      #pragma unroll
      for (int j = 0; j < 3; j++)
        As[s][i * 4 + j] = Kp[i * 3 + 0] * R[0][j] + Kp[i * 3 + 1] * R[1][j] + Kp[i * 3 + 2] * R[2][j];
      As[s][i * 4 + 3] = Kp[i * 3 + 0] * ts[0] + Kp[i * 3 + 1] * ts[1] + Kp[i * 3 + 2] * ts[2];
    }
    float lp_t = -0.5f * (n0 * n0 + n1 * n1 + n2 * n2) - W[31] - C_LP_T;
    float lp_r = -2.f * logf(Mr) - W[32] - C_LP_R;
    As[s][12] = lp_t + lp_r;
    As[s][13] = 0.f; As[s][14] = 0.f; As[s][15] = 0.f;

    // pose_samples: t_out = t_s - R*offset, then q_s
    float* po = out + POSE_OFF + ((size_t)mc * OBJ + o) * 7;
    po[0] = tsx - (R[0][0] * offx + R[0][1] * offy + R[0][2] * offz);
    po[1] = tsy - (R[1][0] * offx + R[1][1] * offy + R[1][2] * offz);
    po[2] = tsz - (R[2][0] * offx + R[2][1] * offy + R[2][2] * offz);
    po[3] = qw; po[4] = qi; po[5] = qj; po[6] = qk;
  }
  __syncthreads();

  const int lane = tid & 31;
  const int wave = tid >> 5;
  const int g = lane >> 4;    // half-wave group
  const int sl = lane & 15;
  const int k0 = 2 * g;       // ISA 32-bit 16x4 A / 4x16 B lane layout: half g holds K=2g,2g+1

  // A fragments (one per output coordinate u, v, z)
  v2f a_u = {As[sl][0 + k0], As[sl][1 + k0]};
  v2f a_v = {As[sl][4 + k0], As[sl][5 + k0]};
  v2f a_z = {As[sl][8 + k0], As[sl][9 + k0]};

  float acc[8];
  #pragma unroll
  for (int r = 0; r < 8; r++) acc[r] = 0.f;

  #pragma unroll
  for (int k = 0; k < 4; ++k) {     // 8 waves x 4 tiles x 16 pts = 512 points
    const int n = (wave * 4 + k) * 16 + sl;
    const float* xp = x3d + ((size_t)o * NPTS + n) * 3;
    float px = xp[0] - offx;
    float py = xp[1] - offy;
    float pz = xp[2] - offz;
    // B fragment: half 0 holds rows K=0,1 (x,y); half 1 holds rows K=2,3 (z,1)
    v2f bf;
    bf.x = g ? pz : px;
    bf.y = g ? 1.0f : py;

    v8f cu = {}, cv = {}, cz = {};
    cu = __builtin_amdgcn_wmma_f32_16x16x4_f32(false, a_u, false, bf, (short)0, cu, false, false);
    cv = __builtin_amdgcn_wmma_f32_16x16x4_f32(false, a_v, false, bf, (short)0, cv, false, false);
    cz = __builtin_amdgcn_wmma_f32_16x16x4_f32(false, a_z, false, bf, (short)0, cz, false, false);

    const float* x2p = x2d + ((size_t)o * NPTS + n) * 2;
    const float* w2p = w2d + ((size_t)o * NPTS + n) * 2;
    const float xu = x2p[0], xv = x2p[1], wu = w2p[0], wv = w2p[1];

    #pragma unroll
    for (int r = 0; r < 8; r++) {
      float z = fmaxf(cz[r], Z_MIN);
      float iz = __builtin_amdgcn_rcpf(z);   // z >= 1e-4, hardware rcp (1 ulp)
      float ru = (cu[r] * iz - xu) * wu;
      float rv = (cv[r] * iz - xv) * wv;
      acc[r] = fmaf(ru, ru, fmaf(rv, rv, acc[r]));
    }
  }

  // deterministic reduction: shfl-xor within each 16-lane half, then LDS across waves
  #pragma unroll
  for (int r = 0; r < 8; r++) {
    float v = acc[r];
    v += __shfl_xor(v, 1, 32);
    v += __shfl_xor(v, 2, 32);
    v += __shfl_xor(v, 4, 32);
    v += __shfl_xor(v, 8, 32);
    acc[r] = v;
  }
  if (sl == 0) {
    #pragma unroll
    for (int r = 0; r < 8; r++) wave_part[wave][r + 8 * g] = acc[r];
  }
  __syncthreads();

  if (tid < 16) {
    float c = 0.f;
    for (int w = 0; w < 8; w++) c += wave_part[w][tid];
    const int mc = tile * 16 + tid;
    out[LW_OFF + (size_t)mc * OBJ + o] = -0.5f * c - As[tid][12];
  }
}

extern "C" void kernel_launch(void* const* d_in, const int* in_sizes, int n_in,
                              void* d_out, int out_size, void* d_ws, size_t ws_size,
                              hipStream_t stream) {
  const float* x3d     = (const float*)d_in[0];
  const float* x2d     = (const float*)d_in[1];
  const float* w2d     = (const float*)d_in[2];
  const float* pose    = (const float*)d_in[3];
  const float* cam     = (const float*)d_in[4];
  const float* cov_t   = (const float*)d_in[5];
  const float* cov_r   = (const float*)d_in[6];
  const float* noise_t = (const float*)d_in[7];
  const float* noise_r = (const float*)d_in[8];
  float* out = (float*)d_out;
  float* ws  = (float*)d_ws;

  epropnp_setup_kernel<<<OBJ, 256, 0, stream>>>(x3d, x2d, w2d, pose, cam, cov_t, cov_r, ws, out);
  epropnp_sample_kernel<<<dim3(MC / 16, OBJ), 256, 0, stream>>>(x3d, x2d, w2d, cam, noise_t, noise_r, ws, out);
}